// RankChannels_59811714564332
// MI455X (gfx1250) — compile-verified
//
#include <hip/hip_runtime.h>
#include <hip/hip_bf16.h>

typedef float v2f __attribute__((ext_vector_type(2)));
typedef float v8f __attribute__((ext_vector_type(8)));

#define CH   1024
#define HW   65536      // 256*256 elements per channel
#define TPB  1024       // 32 waves per block (wave32)

// ---------------------------------------------------------------------------
// Kernel 1: per-channel sum via V_WMMA_F32_16X16X4_F32 with an all-ones B.
// D = A x 1 + C  => every C cell is an accumulator; all 16 columns of D are
// bitwise-identical (same operands, same order), so the cross-lane total is
// exactly 16x the channel sum. Memory-bound: 256 KB per channel streamed with
// b128 loads.
// ---------------------------------------------------------------------------
__global__ __launch_bounds__(TPB)
void RankChannels_mean_kernel(const float* __restrict__ x,
                              float* __restrict__ means) {
    const int c   = blockIdx.x;
    const int tid = threadIdx.x;

    const float4* __restrict__ p =
        (const float4*)(x + (size_t)c * (size_t)HW);   // 16384 float4 / channel

    const v2f ones = {1.0f, 1.0f};
    v8f acc0 = {};
    v8f acc1 = {};

    // 16384 float4 / 1024 threads = 16 iterations, unit-stride coalesced.
    #pragma unroll
    for (int i = 0; i < 16; ++i) {
        float4 v = p[i * TPB + tid];
        v2f a0 = {v.x, v.y};
        v2f a1 = {v.z, v.w};
        // 8 args: (neg_a, A, neg_b, B, c_mod, C, reuse_a, reuse_b)
        acc0 = __builtin_amdgcn_wmma_f32_16x16x4_f32(
                   false, a0, false, ones, (short)0, acc0, false, false);
        acc1 = __builtin_amdgcn_wmma_f32_16x16x4_f32(
                   false, a1, false, ones, (short)0, acc1, false, false);
    }

    // Fold the 16x16 accumulator tile held across the wave.
    float s = (acc0[0] + acc0[1]) + (acc0[2] + acc0[3])
            + (acc0[4] + acc0[5]) + (acc0[6] + acc0[7])
            + (acc1[0] + acc1[1]) + (acc1[2] + acc1[3])
            + (acc1[4] + acc1[5]) + (acc1[6] + acc1[7]);

    // Deterministic LDS tree reduction across the 1024 threads.
    __shared__ float red[TPB];
    red[tid] = s;
    __syncthreads();
    #pragma unroll
    for (int off = TPB / 2; off > 0; off >>= 1) {
        if (tid < off) red[tid] += red[tid + off];
        __syncthreads();
    }

    if (tid == 0) {
        // Cross-lane total = 16 * channel_sum; mean = sum / 65536.
        // 16 * 65536 = 2^20 -> exact power-of-two scale, no rounding.
        means[c] = red[0] * 0x1p-20f;
    }
}

// ---------------------------------------------------------------------------
// Kernel 2: top-256 indices by descending mean, jax.lax.top_k tie semantics
// (equal values ordered by lower index). Rank-based: each slot 0..255 is
// written exactly once -> fully deterministic, no sort needed.
// ---------------------------------------------------------------------------
__global__ __launch_bounds__(CH)
void RankChannels_topk_kernel(const float* __restrict__ means,
                              int* __restrict__ out) {
    __shared__ float m[CH];
    const int tid = threadIdx.x;
    m[tid] = means[tid];
    __syncthreads();

    const float mine = m[tid];
    int rank = 0;
    #pragma unroll 8
    for (int j = 0; j < CH; ++j) {
        float mj = m[j];
        rank += (mj > mine) || ((mj == mine) && (j < tid));
    }
    if (rank < 256) out[rank] = tid;
}

// ---------------------------------------------------------------------------
extern "C" void kernel_launch(void* const* d_in, const int* in_sizes, int n_in,
                              void* d_out, int out_size, void* d_ws, size_t ws_size,
                              hipStream_t stream) {
    (void)in_sizes; (void)n_in; (void)out_size; (void)ws_size;

    const float* x   = (const float*)d_in[0];
    float* means     = (float*)d_ws;          // 1024 floats of scratch
    int*   out_idx   = (int*)d_out;           // 256 int32 channel indices

    RankChannels_mean_kernel<<<CH, TPB, 0, stream>>>(x, means);
    RankChannels_topk_kernel<<<1,  CH,  0, stream>>>(means, out_idx);
}